// DecoupledIKLoss_35905926594716
// MI455X (gfx1250) — compile-verified
//
#include <hip/hip_runtime.h>
#include <hip/hip_bf16.h>

typedef float v2f __attribute__((ext_vector_type(2)));
typedef float v4f __attribute__((ext_vector_type(4)));
typedef float v8f __attribute__((ext_vector_type(8)));

// tanh(x) = 1 - 2/(e^{2x}+1), via v_exp_f32 + v_rcp_f32.
// Saturates exactly to +/-1 for large |x| (inf -> rcp=0 -> 1; 0 -> rcp=1 -> -1).
__device__ __forceinline__ float ftanh(float x) {
  float u = __builtin_amdgcn_exp2f(x * 2.8853900817779268f); // 2*log2(e)
  float r = __builtin_amdgcn_rcpf(u + 1.0f);
  return fmaf(-2.0f, r, 1.0f);
}

// Per-row loss contribution:
//   W_SC * sum_sq_sc/6 + W_WC * sum_sq_wc/3 + W_CIRC * sum_sq_circ/3
// p[6]: raw preds (s0,c0,s1,c1,s2,c2), t[6]: targets, (t5x,t5y,t5z): P5 target.
__device__ __forceinline__ float row_contrib(const float* p, const float* t,
                                             float t5x, float t5y, float t5z) {
  const float d1 = 671.83f, a2 = 431.8f, d2 = 139.7f, a3 = -20.32f, D4 = 431.8f;
  const float inv900 = 1.0f / 900.0f;
  float S[3], C[3];
  float sc = 0.0f, circ = 0.0f;
#pragma unroll
  for (int j = 0; j < 3; ++j) {
    float s = ftanh(p[2 * j]);
    float c = ftanh(p[2 * j + 1]);
    float ds = s - t[2 * j];
    float dc = c - t[2 * j + 1];
    sc = fmaf(ds, ds, sc);
    sc = fmaf(dc, dc, sc);
    float r2 = fmaf(s, s, c * c);           // s^2 + c^2
    float e  = r2 - 1.0f;
    circ = fmaf(e, e, circ);
    // cos(atan2(s,c)) = c/sqrt(s^2+c^2), sin(...) = s/sqrt(...): no trig needed
    float ir = __builtin_amdgcn_rsqf(r2);
    S[j] = s * ir;
    C[j] = c * ir;
  }
  // Closed-form PUMA560 wrist center (alpha = -90, 0, +90 folded in)
  float c23 = fmaf(C[1], C[2], -(S[1] * S[2]));
  float s23 = fmaf(S[1], C[2], C[1] * S[2]);
  float px  = fmaf(a2, C[1], a3 * c23);
  float py  = fmaf(a2, S[1], a3 * s23);
  float w   = fmaf(D4, s23, px);            // px + D4*s23
  float x   = fmaf(C[0], w, -(S[0] * d2));  // c1*w - s1*d2
  float y   = fmaf(S[0], w, C[0] * d2);     // s1*w + c1*d2
  float z   = fmaf(D4, c23, d1 - py);       // d1 - py + D4*c23
  float dx = (x - t5x) * inv900;
  float dy = (y - t5y) * inv900;
  float dz = (z - t5z) * inv900;
  float wc = fmaf(dx, dx, fmaf(dy, dy, dz * dz));
  // weights: W_SC=1 (/6), W_WC=2 (/3), W_CIRC=0.05 (/3)
  return fmaf(sc, (1.0f / 6.0f), fmaf(wc, (2.0f / 3.0f), circ * (0.05f / 3.0f)));
}

// Stage 1: each thread handles a 2-row chunk (48B pred + 48B target + 24B P5,
// all naturally aligned -> b128/b64 non-temporal streaming loads), then an LDS
// tree reduce produces one partial per block.
__global__ __launch_bounds__(256) void ik_partial(const float* __restrict__ pred,
                                                  const float* __restrict__ targ,
                                                  const float* __restrict__ p5t,
                                                  float* __restrict__ partial,
                                                  int nChunks) {
  const int tid = threadIdx.x;
  const int gid = blockIdx.x * blockDim.x + tid;
  const int gstride = gridDim.x * blockDim.x;
  float acc = 0.0f;
  for (int cix = gid; cix < nChunks; cix += gstride) {
    const v4f* P = reinterpret_cast<const v4f*>(pred) + 3 * (size_t)cix;
    const v4f* T = reinterpret_cast<const v4f*>(targ) + 3 * (size_t)cix;
    const v2f* Q = reinterpret_cast<const v2f*>(p5t)  + 3 * (size_t)cix;
    v4f p0 = __builtin_nontemporal_load(P);
    v4f p1 = __builtin_nontemporal_load(P + 1);
    v4f p2 = __builtin_nontemporal_load(P + 2);
    v4f t0 = __builtin_nontemporal_load(T);
    v4f t1 = __builtin_nontemporal_load(T + 1);
    v4f t2 = __builtin_nontemporal_load(T + 2);
    v2f q0 = __builtin_nontemporal_load(Q);
    v2f q1 = __builtin_nontemporal_load(Q + 1);
    v2f q2 = __builtin_nontemporal_load(Q + 2);
    float pr[12] = {p0[0], p0[1], p0[2], p0[3], p1[0], p1[1],
                    p1[2], p1[3], p2[0], p2[1], p2[2], p2[3]};
    float tg[12] = {t0[0], t0[1], t0[2], t0[3], t1[0], t1[1],
                    t1[2], t1[3], t2[0], t2[1], t2[2], t2[3]};
    acc += row_contrib(pr,     tg,     q0[0], q0[1], q1[0]);
    acc += row_contrib(pr + 6, tg + 6, q1[1], q2[0], q2[1]);
  }
  __shared__ float sm[256];
  sm[tid] = acc;
  __syncthreads();
#pragma unroll
  for (int o = 128; o > 0; o >>= 1) {
    if (tid < o) sm[tid] += sm[tid + o];
    __syncthreads();
  }
  if (tid == 0) partial[blockIdx.x] = sm[0];
}

// Stage 2: single wave32. Reduce the block partials 64-at-a-time with
// V_WMMA_F32_16X16X4_F32: A = all-ones(16x4), B = 64 partials, C = accumulator.
// With A==1, every row of D is the vector of column sums, so D row 0
// (c[0], lanes 0..15) carries 16 bins whose sum equals the sum of ALL 64
// packed B values -- independent of the exact B slot<->element mapping.
__global__ __launch_bounds__(32) void ik_final(const float* __restrict__ partial,
                                               float* __restrict__ out,
                                               int n, float invB) {
  const int lane = threadIdx.x; // 0..31, EXEC all ones (required for WMMA)
  v2f ones;
  ones[0] = 1.0f;
  ones[1] = 1.0f;
  v8f c = {};
  int i = 0;
  for (; i + 64 <= n; i += 64) {
    v2f b;
    b[0] = partial[i + lane];
    b[1] = partial[i + 32 + lane];
    c = __builtin_amdgcn_wmma_f32_16x16x4_f32(
        /*neg_a=*/false, ones, /*neg_b=*/false, b,
        /*c_mod=*/(short)0, c, /*reuse_a=*/false, /*reuse_b=*/false);
  }
  float tail = 0.0f;
  for (int j = i + lane; j < n; j += 32) tail += partial[j];
  float v = (lane < 16) ? c[0] : 0.0f; // D row M=0: column sums in lanes 0..15
  v += tail;
#pragma unroll
  for (int off = 16; off > 0; off >>= 1) v += __shfl_xor(v, off, 32);
  if (lane == 0) out[0] = v * invB;
}

extern "C" void kernel_launch(void* const* d_in, const int* in_sizes, int n_in,
                              void* d_out, int out_size, void* d_ws, size_t ws_size,
                              hipStream_t stream) {
  const float* pred = (const float*)d_in[0]; // [B,6]
  const float* targ = (const float*)d_in[1]; // [B,6]
  const float* p5t  = (const float*)d_in[2]; // [B,3]
  float* out = (float*)d_out;

  const long long Bn = (long long)in_sizes[0] / 6; // 4194304
  int nChunks = (int)(Bn / 2);                     // 2 rows per chunk
  const int threads = 256;
  int blocks = (nChunks + threads - 1) / threads;  // 8192 for B=4M (exact fit)
  if (blocks > 8192) blocks = 8192;
  if (blocks < 1) blocks = 1;

  float* partials = (float*)d_ws; // blocks * 4 bytes (<= 32 KB)

  ik_partial<<<blocks, threads, 0, stream>>>(pred, targ, p5t, partials, nChunks);

  const float invB = (float)(1.0 / (double)Bn);
  ik_final<<<1, 32, 0, stream>>>(partials, out, blocks, invB);
}